// HGNN_43224550867516
// MI455X (gfx1250) — compile-verified
//
#include <hip/hip_runtime.h>
#include <hip/hip_bf16.h>

typedef __attribute__((ext_vector_type(2))) float v2f;
typedef __attribute__((ext_vector_type(8))) float v8f;

#define T_PER_B 256

// ---------------------------------------------------------------------------
// Zero fill
// ---------------------------------------------------------------------------
__global__ void k_zero(float* __restrict__ p, int n) {
    int i = blockIdx.x * blockDim.x + threadIdx.x;
    if (i < n) p[i] = 0.0f;
}

// ---------------------------------------------------------------------------
// Degree accumulation: dV[row[e]] += v[e]; dE[col[e]] += v[e]
// ---------------------------------------------------------------------------
__global__ void k_degrees(const int* __restrict__ row, const int* __restrict__ col,
                          const float* __restrict__ val,
                          float* __restrict__ dV, float* __restrict__ dE, int nnz) {
    int i = blockIdx.x * blockDim.x + threadIdx.x;
    if (i < nnz) {
        float v = val[i];
        atomicAdd(&dV[row[i]], v);
        atomicAdd(&dE[col[i]], v);
    }
}

// dv = dV^-1/2, de = dE^-1  (in place)
__global__ void k_recip(float* __restrict__ dv, float* __restrict__ de, int n) {
    int i = blockIdx.x * blockDim.x + threadIdx.x;
    if (i < n) {
        dv[i] = 1.0f / sqrtf(dv[i]);
        de[i] = 1.0f / de[i];
    }
}

// ---------------------------------------------------------------------------
// GEMM1: T1[N,16] = (X[N,512] @ W1^T[512,16] + b1) * dv[:,None]
// One wave per 16-row tile, V_WMMA_F32_16X16X4_F32.
// K-permuted loading: each lane loads a float4 (lo half: k..k+3, hi half:
// k+4..k+7).  WMMA step 0 consumes slots {k,k+1,k+4,k+5}, step 1 consumes
// {k+2,k+3,k+6,k+7}; A and B use the identical slot->k mapping so the
// K-reduction is exact while loads are b128 (one per 2 WMMAs per matrix).
// ---------------------------------------------------------------------------
__global__ __launch_bounds__(T_PER_B) void k_gemm1(
    const float* __restrict__ X, const float* __restrict__ W1,
    const float* __restrict__ b1, const float* __restrict__ dv,
    float* __restrict__ T1, int n) {
    const int lane = threadIdx.x & 31;
    const int wave = threadIdx.x >> 5;
    const int tile = blockIdx.x * (T_PER_B / 32) + wave;
    const int ntiles = (n + 15) >> 4;
    if (tile >= ntiles) return;

    const int nn = lane & 15;           // N index (B col / C col)
    const int hi = (lane >> 4) & 1;     // upper half-wave
    int mrow = tile * 16 + nn;          // A row this lane loads
    if (mrow >= n) mrow = n - 1;        // clamp (n is a multiple of 16 here)

    const float* __restrict__ xa = X  + (size_t)mrow * 512;
    const float* __restrict__ wb = W1 + (size_t)nn   * 512;   // B[k][nn] = W1[nn][k]

    v8f acc = {};
    for (int k = 0; k < 512; k += 8) {
        const int ka = k + hi * 4;
        const float4 av = *(const float4*)(xa + ka);
        const float4 bv = *(const float4*)(wb + ka);
        v2f a0; a0.x = av.x; a0.y = av.y;
        v2f b0; b0.x = bv.x; b0.y = bv.y;
        acc = __builtin_amdgcn_wmma_f32_16x16x4_f32(false, a0, false, b0,
                                                    (short)0, acc, false, false);
        v2f a1; a1.x = av.z; a1.y = av.w;
        v2f b1v; b1v.x = bv.z; b1v.y = bv.w;
        acc = __builtin_amdgcn_wmma_f32_16x16x4_f32(false, a1, false, b1v,
                                                    (short)0, acc, false, false);
    }

    const float bias = b1[nn];
#pragma unroll
    for (int r = 0; r < 8; ++r) {
        const int rowi = tile * 16 + r + hi * 8;
        if (rowi < n)
            T1[(size_t)rowi * 16 + nn] = (acc[r] + bias) * dv[rowi];
    }
}

// ---------------------------------------------------------------------------
// GEMM2: T2[N,64] = (H1[N,16] @ W2^T[16,64] + b2) * dv[:,None]
// One wave per 16-row tile; 4 column tiles of 16; K=16 with the same
// K-permuted b128 loading.  A (2 x float4 per lane) reused across all 4
// column tiles.
// ---------------------------------------------------------------------------
__global__ __launch_bounds__(T_PER_B) void k_gemm2(
    const float* __restrict__ H1, const float* __restrict__ W2,
    const float* __restrict__ b2, const float* __restrict__ dv,
    float* __restrict__ T2, int n) {
    const int lane = threadIdx.x & 31;
    const int wave = threadIdx.x >> 5;
    const int tile = blockIdx.x * (T_PER_B / 32) + wave;
    const int ntiles = (n + 15) >> 4;
    if (tile >= ntiles) return;

    const int nn = lane & 15;
    const int hi = (lane >> 4) & 1;
    int mrow = tile * 16 + nn;
    if (mrow >= n) mrow = n - 1;

    const float* __restrict__ xa = H1 + (size_t)mrow * 16;
    const float4 av0 = *(const float4*)(xa + 0 + hi * 4);   // k 0..7
    const float4 av1 = *(const float4*)(xa + 8 + hi * 4);   // k 8..15

#pragma unroll
    for (int nt = 0; nt < 4; ++nt) {
        const int colb = nt * 16 + nn;
        const float* __restrict__ wb = W2 + (size_t)colb * 16;  // B[k][col] = W2[col][k]
        const float4 bv0 = *(const float4*)(wb + 0 + hi * 4);
        const float4 bv1 = *(const float4*)(wb + 8 + hi * 4);

        v8f acc = {};
        v2f a, b;
        a.x = av0.x; a.y = av0.y; b.x = bv0.x; b.y = bv0.y;
        acc = __builtin_amdgcn_wmma_f32_16x16x4_f32(false, a, false, b,
                                                    (short)0, acc, false, false);
        a.x = av0.z; a.y = av0.w; b.x = bv0.z; b.y = bv0.w;
        acc = __builtin_amdgcn_wmma_f32_16x16x4_f32(false, a, false, b,
                                                    (short)0, acc, false, false);
        a.x = av1.x; a.y = av1.y; b.x = bv1.x; b.y = bv1.y;
        acc = __builtin_amdgcn_wmma_f32_16x16x4_f32(false, a, false, b,
                                                    (short)0, acc, false, false);
        a.x = av1.z; a.y = av1.w; b.x = bv1.z; b.y = bv1.w;
        acc = __builtin_amdgcn_wmma_f32_16x16x4_f32(false, a, false, b,
                                                    (short)0, acc, false, false);

        const float bias = b2[colb];
#pragma unroll
        for (int r = 0; r < 8; ++r) {
            const int rowi = tile * 16 + r + hi * 8;
            if (rowi < n)
                T2[(size_t)rowi * 64 + colb] = (acc[r] + bias) * dv[rowi];
        }
    }
}

// ---------------------------------------------------------------------------
// SpMM scatter, 16-wide rows: dst[di[e],:] += w * src[si[e],:]
// 4 threads per edge, float4 gather (16B aligned), 4 f32 atomics each.
// escale != null folds the D_E^-1 diagonal into the gather (w = v*de[si]).
// ---------------------------------------------------------------------------
__global__ void k_scatter16(const int* __restrict__ si_idx, const int* __restrict__ di_idx,
                            const float* __restrict__ val, const float* __restrict__ src,
                            const float* __restrict__ escale, float* __restrict__ dst,
                            int nnz) {
    const int gid = blockIdx.x * blockDim.x + threadIdx.x;
    if (gid >= nnz * 4) return;
    const int e = gid >> 2;
    const int q = (gid & 3) * 4;
    const int si = si_idx[e];
    const int di = di_idx[e];
    float w = val[e];
    if (escale) w *= escale[si];
    const float4 y = *(const float4*)(src + (size_t)si * 16 + q);
    float* d = dst + (size_t)di * 16 + q;
    atomicAdd(d + 0, w * y.x);
    atomicAdd(d + 1, w * y.y);
    atomicAdd(d + 2, w * y.z);
    atomicAdd(d + 3, w * y.w);
}

// SpMM scatter, 64-wide rows: 16 threads per edge, float4 each.
__global__ void k_scatter64(const int* __restrict__ si_idx, const int* __restrict__ di_idx,
                            const float* __restrict__ val, const float* __restrict__ src,
                            const float* __restrict__ escale, float* __restrict__ dst,
                            int nnz) {
    const int gid = blockIdx.x * blockDim.x + threadIdx.x;
    if (gid >= nnz * 16) return;
    const int e = gid >> 4;
    const int q = (gid & 15) * 4;
    const int si = si_idx[e];
    const int di = di_idx[e];
    float w = val[e];
    if (escale) w *= escale[si];
    const float4 y = *(const float4*)(src + (size_t)si * 64 + q);
    float* d = dst + (size_t)di * 64 + q;
    atomicAdd(d + 0, w * y.x);
    atomicAdd(d + 1, w * y.y);
    atomicAdd(d + 2, w * y.z);
    atomicAdd(d + 3, w * y.w);
}

// H1 = relu(O1 * dv[:,None]), 16 cols
__global__ void k_relu_scale16(const float* __restrict__ O1, const float* __restrict__ dv,
                               float* __restrict__ H1, int n16) {
    const int i = blockIdx.x * blockDim.x + threadIdx.x;
    if (i < n16) H1[i] = fmaxf(O1[i] * dv[i >> 4], 0.0f);
}

// out *= dv[:,None], 64 cols (in place on d_out)
__global__ void k_finalize64(float* __restrict__ out, const float* __restrict__ dv, int n64) {
    const int i = blockIdx.x * blockDim.x + threadIdx.x;
    if (i < n64) out[i] *= dv[i >> 6];
}

// ---------------------------------------------------------------------------
// Host launch
// Inputs: 0=row 1=col 2=values 3=X 4=W1 5=b1 6=W2 7=b2
// ---------------------------------------------------------------------------
extern "C" void kernel_launch(void* const* d_in, const int* in_sizes, int n_in,
                              void* d_out, int out_size, void* d_ws, size_t ws_size,
                              hipStream_t stream) {
    const int*   row = (const int*)d_in[0];
    const int*   col = (const int*)d_in[1];
    const float* val = (const float*)d_in[2];
    const float* X   = (const float*)d_in[3];
    const float* W1  = (const float*)d_in[4];
    const float* b1  = (const float*)d_in[5];
    const float* W2  = (const float*)d_in[6];
    const float* b2  = (const float*)d_in[7];
    float* out = (float*)d_out;

    const int nnz = in_sizes[0];
    const int N   = in_sizes[3] / 512;   // 100000

    // Workspace layout (floats): dv[N] de[N] T1/H1[16N] Z1[16N] O1[16N] T2[64N] Z2[64N]
    float* ws = (float*)d_ws;
    float* dv = ws;
    float* de = ws + (size_t)N;
    float* T1 = ws + (size_t)2  * N;   // reused as H1 after conv1
    float* Z1 = ws + (size_t)18 * N;
    float* O1 = ws + (size_t)34 * N;
    float* T2 = ws + (size_t)50 * N;
    float* Z2 = ws + (size_t)114 * N;

    const int B = T_PER_B;
    auto blocks = [](long long n, int b) { return (int)((n + b - 1) / b); };

    // 1) zero accumulation buffers (dv,de contiguous; Z1,O1 contiguous; Z2; out)
    k_zero<<<blocks(2LL * N, B), B, 0, stream>>>(dv, 2 * N);
    k_zero<<<blocks(32LL * N, B), B, 0, stream>>>(Z1, 32 * N);   // Z1 + O1
    k_zero<<<blocks(64LL * N, B), B, 0, stream>>>(Z2, 64 * N);
    k_zero<<<blocks(64LL * N, B), B, 0, stream>>>(out, 64 * N);

    // 2) degrees, then dv = d_V^-1/2, de = d_E^-1
    k_degrees<<<blocks(nnz, B), B, 0, stream>>>(row, col, val, dv, de, nnz);
    k_recip<<<blocks(N, B), B, 0, stream>>>(dv, de, N);

    // 3) layer 1: T1 = (X@W1^T + b1) * dv   (WMMA f32 16x16x4)
    const int ntiles = (N + 15) / 16;
    k_gemm1<<<blocks(ntiles, B / 32), B, 0, stream>>>(X, W1, b1, dv, T1, N);

    // 4) conv1: Z1 = segsum(v*T1[row] -> col); O1 = segsum(v*de[col]*Z1[col] -> row)
    k_scatter16<<<blocks(4LL * nnz, B), B, 0, stream>>>(row, col, val, T1, nullptr, Z1, nnz);
    k_scatter16<<<blocks(4LL * nnz, B), B, 0, stream>>>(col, row, val, Z1, de, O1, nnz);

    // 5) H1 = relu(O1 * dv)   (written over T1)
    k_relu_scale16<<<blocks(16LL * N, B), B, 0, stream>>>(O1, dv, T1, 16 * N);

    // 6) layer 2: T2 = (H1@W2^T + b2) * dv   (WMMA f32 16x16x4)
    k_gemm2<<<blocks(ntiles, B / 32), B, 0, stream>>>(T1, W2, b2, dv, T2, N);

    // 7) conv2: Z2 = segsum(v*T2[row] -> col); out = segsum(v*de[col]*Z2[col] -> row)
    k_scatter64<<<blocks(16LL * nnz, B), B, 0, stream>>>(row, col, val, T2, nullptr, Z2, nnz);
    k_scatter64<<<blocks(16LL * nnz, B), B, 0, stream>>>(col, row, val, Z2, de, out, nnz);

    // 8) out *= dv
    k_finalize64<<<blocks(64LL * N, B), B, 0, stream>>>(out, dv, 64 * N);
}